// StreamQ_10934986736229
// MI455X (gfx1250) — compile-verified
//
#include <hip/hip_runtime.h>
#include <hip/hip_bf16.h>

// ---------------------------------------------------------------------------
// StreamQ step for B independent agents, fp32, HBM-bandwidth bound (~0.6 GB
// of param/trace streaming => ~25 us at 23.3 TB/s; math is negligible).
//   kA: per-agent fwd(obs,next_obs via V_WMMA_F32_16X16X4_F32) + bwd -> small vectors
//   kB: deterministic partial L1 sums of z_new = gl*z + g
//   kC: per-agent step coefficient + delta output
//   kD: elementwise param/trace update streamed straight into d_out
// ---------------------------------------------------------------------------

#define BB 128
#define DD 512
#define HH 256
#define AA 8
#define PPER 199176            // params per agent (W1+b1+W2+b2+W3+b3)
#define ROWS 398353            // 1 + 2*PPER  (per-agent output row)
#define GLAM 0.792f            // gamma*lambda
#define GAMMA_ 0.99f
#define SLOPE_ 0.01f
#define LNEPS 1e-5f

// per-agent flat param-space piece boundaries (order: W1,b1,W2,b2,W3,b3)
#define E_W1 131072
#define E_B1 131328
#define E_W2 196864
#define E_B2 197120
#define E_W3 199168

#define NBLK 8                 // kB blocks per agent
#define CHUNK 24897            // PPER / NBLK (exact)

typedef __attribute__((ext_vector_type(2))) float v2f;
typedef __attribute__((ext_vector_type(8))) float v8f;

// ---- workspace layout (floats) --------------------------------------------
#define WS_H1    0
#define WS_H2    (BB*HH)
#define WS_DU1   (2*BB*HH)
#define WS_DU2   (3*BB*HH)
#define WS_DELTA (4*BB*HH)
#define WS_COEF  (4*BB*HH + BB)
#define WS_RESET (4*BB*HH + 2*BB)
#define WS_PART  (4*BB*HH + 3*BB)

// ---------------------------------------------------------------------------
__device__ __forceinline__ float block_sum256(float v, float* red) {
#pragma unroll
  for (int off = 16; off > 0; off >>= 1) v += __shfl_down(v, off, 32);
  const int lane = threadIdx.x & 31, wave = threadIdx.x >> 5;
  if (lane == 0) red[wave] = v;
  __syncthreads();
  if (wave == 0) {
    float r = (lane < 8) ? red[lane] : 0.0f;
#pragma unroll
    for (int off = 4; off > 0; off >>= 1) r += __shfl_down(r, off, 32);
    if (lane == 0) red[0] = r;
  }
  __syncthreads();
  float r = red[0];
  __syncthreads();
  return r;
}

// Dual GEMV (two input vectors in B-matrix columns 0/1) over one pass of W.
// sbuf layout: [xa(K) | xb(K) | zeros(K)] -- per-lane B base chosen ONCE, so
// the inner loop is branch-free: 1x global_load_b64 + 1x ds_load_b64 + 1x WMMA.
// A layout: lane L holds A[M=L&15, K=2*(L>>4)+v] (v = VGPR 0/1).
// B layout: lane L holds B[K=2*(L>>4)+v, N=L&15].
// D layout: lane L, VGPR v holds D[M=v+8*(L>>4), N=L&15].
__device__ __forceinline__ void gemv2_wmma(const float* __restrict__ W,
                                           const float* __restrict__ bias,
                                           const float* sbuf,
                                           float* ua, float* ub, int K) {
  const int lane = threadIdx.x & 31, wave = threadIdx.x >> 5;
  const int m  = lane & 15;       // row of A / column of B,D
  const int kh = lane >> 4;       // K half-offset selector
  const int bo = (m == 0) ? 0 : (m == 1) ? K : 2 * K;  // col0 / col1 / zeros
  const float* bsrc = sbuf + bo + 2 * kh;
  for (int rt = wave; rt < HH / 16; rt += 8) {
    const float* Wr = W + (size_t)(rt * 16 + m) * K + 2 * kh;
    v8f c = {0.f, 0.f, 0.f, 0.f, 0.f, 0.f, 0.f, 0.f};
#pragma unroll 4
    for (int kb = 0; kb < K; kb += 4) {
      v2f a  = *(const v2f*)(Wr + kb);     // 8B-aligned global_load_b64
      v2f bv = *(const v2f*)(bsrc + kb);   // 8B-aligned ds_load_b64 (broadcast)
      c = __builtin_amdgcn_wmma_f32_16x16x4_f32(false, a, false, bv, (short)0,
                                                c, false, false);
    }
    if (m < 2) {
      float* dst = (m == 0) ? ua : ub;
#pragma unroll
      for (int v = 0; v < 8; ++v) {
        const int row = rt * 16 + 8 * kh + v;
        dst[row] = c[v] + bias[row];
      }
    }
  }
  __syncthreads();
}

// LayerNorm (no affine) + LeakyReLU in place on u[HH]; returns 1/sigma.
__device__ __forceinline__ float ln_leaky256(float* u, float* red) {
  const int t = threadIdx.x;
  float x = u[t];
  float m = block_sum256(x, red) * (1.0f / HH);
  float d = x - m;
  float var = block_sum256(d * d, red) * (1.0f / HH);
  float s = 1.0f / sqrtf(var + LNEPS);
  float vh = d * s;
  u[t] = (vh >= 0.0f) ? vh : SLOPE_ * vh;
  __syncthreads();
  return s;
}

// ---------------------------------------------------------------------------
__global__ __launch_bounds__(256) void kA(
    const float* __restrict__ W1, const float* __restrict__ b1,
    const float* __restrict__ W2, const float* __restrict__ b2,
    const float* __restrict__ W3, const float* __restrict__ b3,
    const float* __restrict__ obs, const float* __restrict__ nobs,
    const float* __restrict__ reward, const int* __restrict__ action,
    const int* __restrict__ done, const int* __restrict__ explored,
    float* __restrict__ ws) {
  const int b = blockIdx.x;
  const int t = threadIdx.x;
  __shared__ __align__(16) float sb1[3 * DD];   // [obs | next_obs | zeros]
  __shared__ __align__(16) float sb2[3 * HH];   // [h1x | h1n | zeros]
  __shared__ __align__(16) float h2x[HH];
  __shared__ __align__(16) float h2n[HH];
  __shared__ float sdu2[HH];
  __shared__ float red[8];
  __shared__ float sq[2 * AA];
  __shared__ int s_act;

  for (int i = t; i < DD; i += 256) {
    sb1[i] = obs[b * DD + i];
    sb1[DD + i] = nobs[b * DD + i];
    sb1[2 * DD + i] = 0.0f;               // zero pad for WMMA B columns 2..15
  }
  sb2[2 * HH + t] = 0.0f;
  __syncthreads();

  float* h1x = sb2;                       // layer-2 B-buffer slices
  float* h1n = sb2 + HH;

  // layer 1 (obs & next_obs share one pass over W1)
  gemv2_wmma(W1 + (size_t)b * HH * DD, b1 + b * HH, sb1, h1x, h1n, DD);
  float s1 = ln_leaky256(h1x, red);
  ln_leaky256(h1n, red);
  // layer 2
  gemv2_wmma(W2 + (size_t)b * HH * HH, b2 + b * HH, sb2, h2x, h2n, HH);
  float s2 = ln_leaky256(h2x, red);
  ln_leaky256(h2n, red);

  // layer 3: wave w computes q[w] for both states
  {
    const int lane = t & 31, wave = t >> 5;
    const float* Wr = W3 + (size_t)b * AA * HH + wave * HH;
    float ax = 0.f, an = 0.f;
    for (int j = lane; j < HH; j += 32) {
      float w = Wr[j];
      ax += w * h2x[j];
      an += w * h2n[j];
    }
#pragma unroll
    for (int off = 16; off > 0; off >>= 1) {
      ax += __shfl_down(ax, off, 32);
      an += __shfl_down(an, off, 32);
    }
    if (lane == 0) {
      float bias = b3[b * AA + wave];
      sq[wave] = ax + bias;
      sq[AA + wave] = an + bias;
    }
  }
  __syncthreads();
  if (t == 0) {
    float qmax = sq[AA];
#pragma unroll
    for (int a = 1; a < AA; ++a) qmax = fmaxf(qmax, sq[AA + a]);
    const int act = action[b];
    const float df = (float)done[b];
    const float delta = reward[b] + (1.0f - df) * GAMMA_ * qmax - sq[act];
    ws[WS_DELTA + b] = delta;
    ws[WS_RESET + b] = (done[b] || explored[b]) ? 1.0f : 0.0f;
    s_act = act;
  }
  __syncthreads();
  const int act = s_act;

  // backward through layer 2 (gradient of -Q(obs)[act])
  float dh2 = -W3[(size_t)b * AA * HH + act * HH + t];
  float h2v = h2x[t];
  float v2h = (h2v >= 0.f) ? h2v : h2v * (1.0f / SLOPE_);  // recover LN output
  float dvh2 = (h2v >= 0.f) ? dh2 : SLOPE_ * dh2;
  float mdv2 = block_sum256(dvh2, red) * (1.0f / HH);
  float mdvx2 = block_sum256(dvh2 * v2h, red) * (1.0f / HH);
  float du2 = s2 * (dvh2 - mdv2 - v2h * mdvx2);
  sdu2[t] = du2;
  __syncthreads();

  // dh1 = W2^T @ du2 (coalesced column walk)
  float acc = 0.f;
  const float* W2b = W2 + (size_t)b * HH * HH;
#pragma unroll 4
  for (int i = 0; i < HH; ++i) acc += W2b[(size_t)i * HH + t] * sdu2[i];
  float dh1 = acc;

  float h1v = h1x[t];
  float v1h = (h1v >= 0.f) ? h1v : h1v * (1.0f / SLOPE_);
  float dvh1 = (h1v >= 0.f) ? dh1 : SLOPE_ * dh1;
  float mdv1 = block_sum256(dvh1, red) * (1.0f / HH);
  float mdvx1 = block_sum256(dvh1 * v1h, red) * (1.0f / HH);
  float du1 = s1 * (dvh1 - mdv1 - v1h * mdvx1);

  ws[WS_H1 + b * HH + t] = h1v;
  ws[WS_H2 + b * HH + t] = h2v;
  ws[WS_DU1 + b * HH + t] = du1;
  ws[WS_DU2 + b * HH + t] = du2;
}

// ---------------------------------------------------------------------------
// decode one per-agent flat param index -> (z value, gradient value)
__device__ __forceinline__ void decode_zg(
    int b, int idx, int act,
    const float* zW1, const float* zb1, const float* zW2, const float* zb2,
    const float* zW3, const float* zb3, const float* obs, const float* ws,
    float& z, float& g) {
  if (idx < E_W1) {
    int i = idx >> 9, j = idx & 511;
    z = zW1[(size_t)b * E_W1 + idx];
    g = ws[WS_DU1 + b * HH + i] * obs[b * DD + j];
  } else if (idx < E_B1) {
    int i = idx - E_W1;
    z = zb1[b * HH + i];
    g = ws[WS_DU1 + b * HH + i];
  } else if (idx < E_W2) {
    int r = idx - E_B1;
    int i = r >> 8, j = r & 255;
    z = zW2[(size_t)b * (HH * HH) + r];
    g = ws[WS_DU2 + b * HH + i] * ws[WS_H1 + b * HH + j];
  } else if (idx < E_B2) {
    int i = idx - E_W2;
    z = zb2[b * HH + i];
    g = ws[WS_DU2 + b * HH + i];
  } else if (idx < E_W3) {
    int r = idx - E_B2;
    int a = r >> 8, j = r & 255;
    z = zW3[(size_t)b * (AA * HH) + r];
    g = (a == act) ? -ws[WS_H2 + b * HH + j] : 0.f;
  } else {
    int a = idx - E_W3;
    z = zb3[b * AA + a];
    g = (a == act) ? -1.f : 0.f;
  }
}

__global__ __launch_bounds__(256) void kB(
    const float* __restrict__ zW1, const float* __restrict__ zb1,
    const float* __restrict__ zW2, const float* __restrict__ zb2,
    const float* __restrict__ zW3, const float* __restrict__ zb3,
    const float* __restrict__ obs, const int* __restrict__ action,
    const float* __restrict__ ws, float* __restrict__ partials) {
  const int b = blockIdx.x >> 3;
  const int k = blockIdx.x & 7;
  const int lo = k * CHUNK;
  const int hi = min(lo + CHUNK, PPER);
  const int act = action[b];
  float acc = 0.f;
  for (int idx = lo + (int)threadIdx.x; idx < hi; idx += 256) {
    float z, g;
    decode_zg(b, idx, act, zW1, zb1, zW2, zb2, zW3, zb3, obs, ws, z, g);
    acc += fabsf(GLAM * z + g);
  }
  __shared__ float red[8];
#pragma unroll
  for (int off = 16; off > 0; off >>= 1) acc += __shfl_down(acc, off, 32);
  const int lane = threadIdx.x & 31, wave = threadIdx.x >> 5;
  if (lane == 0) red[wave] = acc;
  __syncthreads();
  if (threadIdx.x == 0) {
    float r = 0.f;
#pragma unroll
    for (int w = 0; w < 8; ++w) r += red[w];
    partials[blockIdx.x] = r;   // fixed order => deterministic
  }
}

// ---------------------------------------------------------------------------
__global__ void kC(float* __restrict__ ws, float* __restrict__ out) {
  const int b = threadIdx.x;
  if (b < BB) {
    float zs = 0.f;
#pragma unroll
    for (int k = 0; k < NBLK; ++k) zs += ws[WS_PART + b * NBLK + k];
    const float delta = ws[WS_DELTA + b];
    const float db = fmaxf(fabsf(delta), 1.0f);
    const float dot = 2.0f * db * zs;              // ALPHA*KAPPA*|d|*zsum
    const float step = 1.0f / fmaxf(dot, 1.0f);    // ALPHA / max(dot,1)
    ws[WS_COEF + b] = step * delta;
    out[(size_t)b * ROWS] = delta;
  }
}

// ---------------------------------------------------------------------------
__global__ __launch_bounds__(256) void kD(
    const float* __restrict__ W1, const float* __restrict__ b1,
    const float* __restrict__ W2, const float* __restrict__ b2,
    const float* __restrict__ W3, const float* __restrict__ b3,
    const float* __restrict__ zW1, const float* __restrict__ zb1,
    const float* __restrict__ zW2, const float* __restrict__ zb2,
    const float* __restrict__ zW3, const float* __restrict__ zb3,
    const float* __restrict__ obs, const int* __restrict__ action,
    const float* __restrict__ ws, float* __restrict__ out) {
  const int gid = blockIdx.x * 256 + threadIdx.x;
  if (gid >= BB * PPER) return;
  const int b = gid / PPER;
  const int idx = gid - b * PPER;
  const int act = action[b];
  const float coef = ws[WS_COEF + b];
  const float keep = (ws[WS_RESET + b] != 0.0f) ? 0.0f : 1.0f;

  float z, g;
  decode_zg(b, idx, act, zW1, zb1, zW2, zb2, zW3, zb3, obs, ws, z, g);

  float p;
  if (idx < E_W1)      p = W1[(size_t)b * E_W1 + idx];
  else if (idx < E_B1) p = b1[b * HH + (idx - E_W1)];
  else if (idx < E_W2) p = W2[(size_t)b * (HH * HH) + (idx - E_B1)];
  else if (idx < E_B2) p = b2[b * HH + (idx - E_W2)];
  else if (idx < E_W3) p = W3[(size_t)b * (AA * HH) + (idx - E_B2)];
  else                 p = b3[b * AA + (idx - E_W3)];

  const float zn = GLAM * z + g;
  const size_t base = (size_t)b * ROWS;
  out[base + 1 + idx] = p - coef * zn;            // new_params region
  out[base + 1 + PPER + idx] = keep * zn;         // z_out region
}

// ---------------------------------------------------------------------------
extern "C" void kernel_launch(void* const* d_in, const int* in_sizes, int n_in,
                              void* d_out, int out_size, void* d_ws,
                              size_t ws_size, hipStream_t stream) {
  const float* W1 = (const float*)d_in[0];
  const float* b1 = (const float*)d_in[1];
  const float* W2 = (const float*)d_in[2];
  const float* b2 = (const float*)d_in[3];
  const float* W3 = (const float*)d_in[4];
  const float* b3 = (const float*)d_in[5];
  const float* zW1 = (const float*)d_in[6];
  const float* zb1 = (const float*)d_in[7];
  const float* zW2 = (const float*)d_in[8];
  const float* zb2 = (const float*)d_in[9];
  const float* zW3 = (const float*)d_in[10];
  const float* zb3 = (const float*)d_in[11];
  const float* obs = (const float*)d_in[12];
  const float* nobs = (const float*)d_in[13];
  const float* reward = (const float*)d_in[14];
  const int* action = (const int*)d_in[15];
  const int* done = (const int*)d_in[16];
  const int* explored = (const int*)d_in[17];
  float* out = (float*)d_out;
  float* ws = (float*)d_ws;
  float* partials = ws + WS_PART;

  kA<<<BB, 256, 0, stream>>>(W1, b1, W2, b2, W3, b3, obs, nobs, reward, action,
                             done, explored, ws);
  kB<<<BB * NBLK, 256, 0, stream>>>(zW1, zb1, zW2, zb2, zW3, zb3, obs, action,
                                    ws, partials);
  kC<<<1, 128, 0, stream>>>(ws, out);
  const int nblk = (BB * PPER + 255) / 256;
  kD<<<nblk, 256, 0, stream>>>(W1, b1, W2, b2, W3, b3, zW1, zb1, zW2, zb2, zW3,
                               zb3, obs, action, ws, out);
}